// WaveletNeuralStyleTransfer_77395310674224
// MI455X (gfx1250) — compile-verified
//
#include <hip/hip_runtime.h>
#include <hip/hip_bf16.h>
#include <stdint.h>

// Shapes fixed by the reference: (2, 64, 512, 512) f32 -> 128 rows of N = 262144.
#define ROW_N      262144
#define ROW_SHIFT  18
#define SORT_T     256          // threads per sort workgroup
#define SEG        (ROW_N / SORT_T)

// cos constants
#define COS_02PI 0.8090169943749475f   // cos(0.2*pi)  (approx band)
#define COS_03PI 0.5877852522924731f   // cos(0.3*pi)  (detail bands)

typedef unsigned int v4u __attribute__((ext_vector_type(4)));
typedef int          v4i __attribute__((ext_vector_type(4)));
typedef int          v8i __attribute__((ext_vector_type(8)));

// ---------------------------------------------------------------------------
// Pack keys (vectorized b128 streams):
//   srcA = (|c| bits << 32) | idx-in-row ; tgtA = |s| bits.
// Non-negative float bit patterns sort correctly as unsigned integers.
// ---------------------------------------------------------------------------
__global__ void pack_keys(const float* __restrict__ c, const float* __restrict__ s,
                          unsigned long long* __restrict__ srcA,
                          unsigned* __restrict__ tgtA, int total) {
  const int total4 = total >> 2;
  const float4* __restrict__ c4 = (const float4*)c;
  const float4* __restrict__ s4 = (const float4*)s;
  uint4* __restrict__ t4 = (uint4*)tgtA;
  ulonglong2* __restrict__ p2 = (ulonglong2*)srcA;

  for (long long i = (long long)blockIdx.x * blockDim.x + threadIdx.x; i < total4;
       i += (long long)gridDim.x * blockDim.x) {
    float4 cv = c4[i];
    float4 sv = s4[i];
    long long b = i << 2;
    unsigned idx0 = (unsigned)(b & (ROW_N - 1));

    uint4 t;
    t.x = __float_as_uint(fabsf(sv.x));
    t.y = __float_as_uint(fabsf(sv.y));
    t.z = __float_as_uint(fabsf(sv.z));
    t.w = __float_as_uint(fabsf(sv.w));
    t4[i] = t;

    ulonglong2 p0, p1;
    p0.x = ((unsigned long long)__float_as_uint(fabsf(cv.x)) << 32) | (idx0 + 0u);
    p0.y = ((unsigned long long)__float_as_uint(fabsf(cv.y)) << 32) | (idx0 + 1u);
    p1.x = ((unsigned long long)__float_as_uint(fabsf(cv.z)) << 32) | (idx0 + 2u);
    p1.y = ((unsigned long long)__float_as_uint(fabsf(cv.w)) << 32) | (idx0 + 3u);
    p2[2 * i + 0] = p0;
    p2[2 * i + 1] = p1;
  }
}

// ---------------------------------------------------------------------------
// Per-row stable LSD radix sort (4-bit digits, 8 passes over 32 key bits).
// One workgroup of 256 threads owns one row; each thread owns a contiguous
// segment of SEG elements (b128 vector reads), so the column scan over
// per-thread LDS histograms yields a stable scatter. Data ends back in A.
// ---------------------------------------------------------------------------
__global__ void radix_sort_rows_u64(unsigned long long* __restrict__ A,
                                    unsigned long long* __restrict__ B) {
  __shared__ unsigned lhist[SORT_T][16];
  __shared__ unsigned digTotal[16];
  __shared__ unsigned digBase[16];

  const int tid = threadIdx.x;
  const long long rowOff = (long long)blockIdx.x << ROW_SHIFT;
  unsigned long long* buf0 = A + rowOff;
  unsigned long long* buf1 = B + rowOff;
  const int s0 = tid * SEG;

  for (int pass = 0; pass < 8; ++pass) {
    const int shift = 32 + pass * 4;
    unsigned long long* __restrict__ src = (pass & 1) ? buf1 : buf0;
    unsigned long long* __restrict__ dst = (pass & 1) ? buf0 : buf1;
    const ulonglong2* __restrict__ src2 = (const ulonglong2*)(src + s0);

#pragma unroll
    for (int d = 0; d < 16; ++d) lhist[tid][d] = 0;
    __builtin_prefetch(src + s0, 0, 0);            // gfx1250 global_prefetch
    for (int i = 0; i < SEG / 2; ++i) {
      ulonglong2 v = src2[i];
      lhist[tid][(unsigned)((v.x >> shift) & 15ull)]++;
      lhist[tid][(unsigned)((v.y >> shift) & 15ull)]++;
    }
    __syncthreads();

    if (tid < 16) {
      unsigned acc = 0;
      for (int t = 0; t < SORT_T; ++t) {
        unsigned x = lhist[t][tid];
        lhist[t][tid] = acc;        // exclusive prefix within digit column
        acc += x;
      }
      digTotal[tid] = acc;
    }
    __syncthreads();
    if (tid == 0) {
      unsigned acc = 0;
#pragma unroll
      for (int d = 0; d < 16; ++d) { digBase[d] = acc; acc += digTotal[d]; }
    }
    __syncthreads();

    for (int i = 0; i < SEG / 2; ++i) {
      ulonglong2 v = src2[i];
      unsigned d0 = (unsigned)((v.x >> shift) & 15ull);
      dst[digBase[d0] + lhist[tid][d0]++] = v.x;
      unsigned d1 = (unsigned)((v.y >> shift) & 15ull);
      dst[digBase[d1] + lhist[tid][d1]++] = v.y;
    }
    __syncthreads();
  }
}

__global__ void radix_sort_rows_u32(unsigned* __restrict__ A, unsigned* __restrict__ B) {
  __shared__ unsigned lhist[SORT_T][16];
  __shared__ unsigned digTotal[16];
  __shared__ unsigned digBase[16];

  const int tid = threadIdx.x;
  const long long rowOff = (long long)blockIdx.x << ROW_SHIFT;
  unsigned* buf0 = A + rowOff;
  unsigned* buf1 = B + rowOff;
  const int s0 = tid * SEG;

  for (int pass = 0; pass < 8; ++pass) {
    const int shift = pass * 4;
    unsigned* __restrict__ src = (pass & 1) ? buf1 : buf0;
    unsigned* __restrict__ dst = (pass & 1) ? buf0 : buf1;
    const uint4* __restrict__ src4 = (const uint4*)(src + s0);

#pragma unroll
    for (int d = 0; d < 16; ++d) lhist[tid][d] = 0;
    __builtin_prefetch(src + s0, 0, 0);
    for (int i = 0; i < SEG / 4; ++i) {
      uint4 v = src4[i];
      lhist[tid][(v.x >> shift) & 15u]++;
      lhist[tid][(v.y >> shift) & 15u]++;
      lhist[tid][(v.z >> shift) & 15u]++;
      lhist[tid][(v.w >> shift) & 15u]++;
    }
    __syncthreads();

    if (tid < 16) {
      unsigned acc = 0;
      for (int t = 0; t < SORT_T; ++t) {
        unsigned x = lhist[t][tid];
        lhist[t][tid] = acc;
        acc += x;
      }
      digTotal[tid] = acc;
    }
    __syncthreads();
    if (tid == 0) {
      unsigned acc = 0;
#pragma unroll
      for (int d = 0; d < 16; ++d) { digBase[d] = acc; acc += digTotal[d]; }
    }
    __syncthreads();

    for (int i = 0; i < SEG / 4; ++i) {
      uint4 v = src4[i];
      unsigned d0 = (v.x >> shift) & 15u; dst[digBase[d0] + lhist[tid][d0]++] = v.x;
      unsigned d1 = (v.y >> shift) & 15u; dst[digBase[d1] + lhist[tid][d1]++] = v.y;
      unsigned d2 = (v.z >> shift) & 15u; dst[digBase[d2] + lhist[tid][d2]++] = v.z;
      unsigned d3 = (v.w >> shift) & 15u; dst[digBase[d3] + lhist[tid][d3]++] = v.w;
    }
    __syncthreads();
  }
}

// ---------------------------------------------------------------------------
// Approx band: out[row, idx(k)] = style_sorted[row, k] * cosfactor(signs).
// cos(0.8*phase_c + 0.2*phase_s), phases in {0, pi}.
// ---------------------------------------------------------------------------
__global__ void approx_map(const unsigned long long* __restrict__ srcSorted,
                           const unsigned* __restrict__ tgtSorted,
                           const float* __restrict__ c, const float* __restrict__ s,
                           float* __restrict__ out, int total) {
  for (long long i = (long long)blockIdx.x * blockDim.x + threadIdx.x; i < total;
       i += (long long)gridDim.x * blockDim.x) {
    long long row = i >> ROW_SHIFT;
    unsigned long long v = srcSorted[i];
    unsigned idx = (unsigned)(v & 0xFFFFFFFFull);      // original in-row index
    long long g = (row << ROW_SHIFT) + (long long)idx;
    float tval = __uint_as_float(tgtSorted[i]);        // matched magnitude
    float cv = c[g], sv = s[g];
    float f = (cv < 0.0f) ? ((sv < 0.0f) ? -1.0f : -COS_02PI)
                          : ((sv < 0.0f) ?  COS_02PI :  1.0f);
    out[g] = tval * f;
  }
}

// ---------------------------------------------------------------------------
// sum(|x|) stage 1: tiles double-buffered through LDS via CDNA5 async
// global->LDS loads (ASYNCcnt). Issue tile k+1, wait asynccnt<=1 (in-order
// completion => tile k resident), consume, flip. s_wait_dscnt 0 protects the
// LDS overwrite against in-flight ds reads from the previous iteration.
// ---------------------------------------------------------------------------
__global__ void reduce_abs(const float* __restrict__ x, float* __restrict__ partial,
                           int n) {
  __shared__ __align__(16) float tile[2][1024];
  __shared__ float sm[256];
  const int tid = threadIdx.x;
  unsigned lds0 = (unsigned)(uintptr_t)(&tile[0][tid * 4]);
  unsigned lds1 = (unsigned)(uintptr_t)(&tile[1][tid * 4]);
  const long long stride = (long long)gridDim.x * 1024;
  long long base = (long long)blockIdx.x * 1024;
  float acc = 0.0f;

  if (base < n) {
    asm volatile("global_load_async_to_lds_b128 %0, %1, off"
                 :: "v"(lds0), "v"(x + base + tid * 4) : "memory");
  }
  int buf = 0;
  for (; base < n; base += stride, buf ^= 1) {
    long long nxt = base + stride;
    if (nxt < n) {
      asm volatile("s_wait_dscnt 0\n\t"
                   "global_load_async_to_lds_b128 %0, %1, off"
                   :: "v"(buf ? lds0 : lds1), "v"(x + nxt + tid * 4) : "memory");
      asm volatile("s_wait_asynccnt 0x1" ::: "memory");
    } else {
      asm volatile("s_wait_asynccnt 0x0" ::: "memory");
    }
#pragma unroll
    for (int k = 0; k < 4; ++k) acc += fabsf(tile[buf][tid * 4 + k]);
  }

  sm[tid] = acc;
  __syncthreads();
  for (int off = 128; off > 0; off >>= 1) {
    if (tid < off) sm[tid] += sm[tid + off];
    __syncthreads();
  }
  if (tid == 0) partial[blockIdx.x] = sm[0];
}

__global__ void finalize_sums(const float* __restrict__ partials,
                              float* __restrict__ sums) {
  __shared__ float sm[256];
  for (int j = 0; j < 6; ++j) {
    float acc = 0.0f;
    for (int i = threadIdx.x; i < 1024; i += 256) acc += partials[j * 1024 + i];
    sm[threadIdx.x] = acc;
    __syncthreads();
    for (int off = 128; off > 0; off >>= 1) {
      if (threadIdx.x < off) sm[threadIdx.x] += sm[threadIdx.x + off];
      __syncthreads();
    }
    if (threadIdx.x == 0) sums[j] = sm[0];
    __syncthreads();
  }
}

// ---------------------------------------------------------------------------
// TDM helper: issue a 1024-float contiguous tile load Global->LDS via the
// Tensor Data Mover (TENSORcnt). D# per CDNA5 ISA:
//   group0: count=1 | lds_addr | global_addr[56:0] | type=2
//   group1: data_size=4B, tensor_dim0=1024, tensor_dim1=1, tile_dim0=1024,
//           tile_dim1=1, tensor_dim0_stride=1024
// ---------------------------------------------------------------------------
__device__ __forceinline__ void tdm_load_tile_1k(unsigned lds_addr,
                                                 const float* gptr) {
  unsigned long long ga = (unsigned long long)(uintptr_t)gptr;
  v4u g0 = { 1u,                               // count=1, user mode
             lds_addr,                         // LDS byte address
             (unsigned)ga,                     // global_addr[31:0]
             (((unsigned)(ga >> 32)) & 0x01FFFFFFu) | 0x80000000u }; // [56:32] | type=2
  v8i g1 = { 0x00020000,                       // data_size=2 (4 bytes)
             0x04000000,                       // tensor_dim0[15:0]=1024 in [31:16]
             0x00010000,                       // tensor_dim0[31:16]=0, tensor_dim1[15:0]=1
             0x04000000,                       // tensor_dim1[31:16]=0, tile_dim0=1024
             0x00000001,                       // tile_dim1=1, tile_dim2=0
             0x00000400,                       // tensor_dim0_stride[31:0]=1024
             0x00000000,                       // stride[47:32]=0, dim1_stride[15:0]=0
             0x00000000 };                     // tensor_dim1_stride[47:16]=0
  v4i z4 = { 0, 0, 0, 0 };
#if __clang_major__ >= 23
  v8i z8 = { 0, 0, 0, 0, 0, 0, 0, 0 };
  __builtin_amdgcn_tensor_load_to_lds(g0, g1, z4, z4, z8, 0);
#else
  __builtin_amdgcn_tensor_load_to_lds(g0, g1, z4, z4, 0);
#endif
}

// ---------------------------------------------------------------------------
// Detail bands: out = |c| * 1.8 * (sum|s|/sum|c|) * cos(0.7*ph_c + 0.3*ph_s).
// One 4KB tile per input per block, DMA'd into LDS by the TDM; wave 0 issues
// (TDM ignores EXEC -> exactly one issuing wave), waits TENSORcnt, barrier.
// ---------------------------------------------------------------------------
__global__ void detail_map(const float* __restrict__ c, const float* __restrict__ s,
                           const float* __restrict__ sums, int ci, int si,
                           float* __restrict__ out, int n) {
  __shared__ __align__(16) float c_tile[1024];
  __shared__ __align__(16) float s_tile[1024];
  const int tid = threadIdx.x;
  const long long base = (long long)blockIdx.x * 1024;   // one full tile per block

  if (tid < 32) {                                        // wave 0 only (wave32)
    tdm_load_tile_1k((unsigned)(uintptr_t)c_tile, c + base);
    tdm_load_tile_1k((unsigned)(uintptr_t)s_tile, s + base);
    __builtin_amdgcn_s_wait_tensorcnt(0);
  }
  __syncthreads();

  const float ratio = 1.8f * sums[si] / sums[ci];
  float4 o;
#pragma unroll
  for (int k = 0; k < 4; ++k) {
    float cv = c_tile[tid * 4 + k];
    float sv = s_tile[tid * 4 + k];
    float f = (cv < 0.0f) ? ((sv < 0.0f) ? -1.0f : -COS_03PI)
                          : ((sv < 0.0f) ?  COS_03PI :  1.0f);
    float r = fabsf(cv) * ratio * f;
    if (k == 0) o.x = r; else if (k == 1) o.y = r; else if (k == 2) o.z = r; else o.w = r;
  }
  ((float4*)(out + base))[tid] = o;
}

// ---------------------------------------------------------------------------
// Launch
// ---------------------------------------------------------------------------
extern "C" void kernel_launch(void* const* d_in, const int* in_sizes, int n_in,
                              void* d_out, int out_size, void* d_ws, size_t ws_size,
                              hipStream_t stream) {
  const float* c_ap = (const float*)d_in[0];
  const float* s_ap = (const float*)d_in[1];
  const float* c_h  = (const float*)d_in[2];
  const float* c_v  = (const float*)d_in[3];
  const float* c_dd = (const float*)d_in[4];
  const float* s_h  = (const float*)d_in[5];
  const float* s_v  = (const float*)d_in[6];
  const float* s_dd = (const float*)d_in[7];

  const int total = in_sizes[0];          // 33554432
  const int R = total / ROW_N;            // 128 rows

  float* out = (float*)d_out;

  // workspace layout
  float* sums     = (float*)d_ws;                         // 6 floats
  float* partials = (float*)d_ws + 16;                    // 6*1024 floats
  unsigned long long* srcA = (unsigned long long*)((char*)d_ws + 65536);
  unsigned long long* srcB = srcA + (size_t)total;
  unsigned* tgtA = (unsigned*)(srcB + (size_t)total);
  unsigned* tgtB = tgtA + (size_t)total;

  const int EW_BLOCKS = 4096;

  // ---- approximation band: pack -> sort source(u64) + target(u32) -> map ----
  pack_keys<<<EW_BLOCKS, 256, 0, stream>>>(c_ap, s_ap, srcA, tgtA, total);
  radix_sort_rows_u64<<<R, SORT_T, 0, stream>>>(srcA, srcB);
  radix_sort_rows_u32<<<R, SORT_T, 0, stream>>>(tgtA, tgtB);
  approx_map<<<EW_BLOCKS, 256, 0, stream>>>(srcA, tgtA, c_ap, s_ap, out, total);

  // ---- detail bands: deterministic two-stage reductions, then elementwise ----
  reduce_abs<<<1024, 256, 0, stream>>>(c_h,  partials + 0 * 1024, total);
  reduce_abs<<<1024, 256, 0, stream>>>(s_h,  partials + 1 * 1024, total);
  reduce_abs<<<1024, 256, 0, stream>>>(c_v,  partials + 2 * 1024, total);
  reduce_abs<<<1024, 256, 0, stream>>>(s_v,  partials + 3 * 1024, total);
  reduce_abs<<<1024, 256, 0, stream>>>(c_dd, partials + 4 * 1024, total);
  reduce_abs<<<1024, 256, 0, stream>>>(s_dd, partials + 5 * 1024, total);
  finalize_sums<<<1, 256, 0, stream>>>(partials, sums);

  const int DM_BLOCKS = total / 1024;     // full tiles, n % 1024 == 0
  detail_map<<<DM_BLOCKS, 256, 0, stream>>>(c_h,  s_h,  sums, 0, 1,
                                            out + (size_t)total, total);
  detail_map<<<DM_BLOCKS, 256, 0, stream>>>(c_v,  s_v,  sums, 2, 3,
                                            out + 2 * (size_t)total, total);
  detail_map<<<DM_BLOCKS, 256, 0, stream>>>(c_dd, s_dd, sums, 4, 5,
                                            out + 3 * (size_t)total, total);
}